// ClusterAggregator_54039278518791
// MI455X (gfx1250) — compile-verified
//
#include <hip/hip_runtime.h>
#include <math.h>

typedef __attribute__((ext_vector_type(2))) float v2f;
typedef __attribute__((ext_vector_type(8))) float v8f;

#define B_   8
#define N_   32768
#define D_   128
#define H_   64
#define C_   64
#define BN_  (B_ * N_)

// Transposed W1 in LDS: w1t[n][k], padded row stride.
// stride=132 (even): ds_load_b64 stays 8B-aligned; bank = (4n+kk)%64 is
// conflict-free across the whole wave for the WMMA B-fragment reads.
#define W1T_STRIDE 132

// ---------------------------------------------------------------------------
// Kernel 0: zero output + per-segment max/denom accumulators
// ---------------------------------------------------------------------------
__global__ void init_kernel(float* __restrict__ out,
                            float* __restrict__ segmax,
                            float* __restrict__ denom) {
    int i = blockIdx.x * 256 + threadIdx.x;
    if (i < B_ * C_ * D_) out[i] = 0.0f;
    if (i < B_ * C_) { segmax[i] = 0.0f; denom[i] = 0.0f; }
}

// ---------------------------------------------------------------------------
// Kernel 1: importance MLP via fp32 WMMA.
// One wave (32 lanes) owns a 16-token tile: h = relu(A(16x128) * W1(128x64) + b1),
// imp = sigmoid(h * W2 + b2). Also atomic-max the per-segment importance.
// ---------------------------------------------------------------------------
__global__ __launch_bounds__(128) void imp_kernel(
    const float* __restrict__ feat, const int* __restrict__ ca,
    const float* __restrict__ W1,   const float* __restrict__ b1,
    const float* __restrict__ W2,   const float* __restrict__ b2,
    float* __restrict__ imp, unsigned int* __restrict__ segmaxU)
{
    __shared__ float w1t[H_ * W1T_STRIDE];   // 64 * 132 * 4 = 33792 B

    // Stage W1 (128x64 row-major) transposed into LDS: w1t[n*stride + k] = W1[k][n]
    for (int i = threadIdx.x; i < D_ * H_; i += 128) {
        int kk = i >> 6;          // k row
        int n  = i & 63;          // h col
        w1t[n * W1T_STRIDE + kk] = W1[i];
    }
    __syncthreads();

    const int lane = threadIdx.x & 31;
    const int wave = threadIdx.x >> 5;
    const int l15  = lane & 15;
    const int hi   = lane >> 4;                  // 0 or 1
    const int t0   = (blockIdx.x * 4 + wave) * 16;  // first token of this tile

    v8f acc[4];
    #pragma unroll
    for (int nt = 0; nt < 4; ++nt)
        #pragma unroll
        for (int r = 0; r < 8; ++r) acc[nt][r] = 0.0f;

    // A-fragment base: lane holds A[M=l15][K = k+2*hi .. k+2*hi+1]  (8B aligned)
    const float* arow = feat + (size_t)(t0 + l15) * D_ + 2 * hi;

    // B-fragment bases (one per n-tile): lane's column n = nt*16 + l15, offset kk
    const float* brow0 = &w1t[(0 * 16 + l15) * W1T_STRIDE + 2 * hi];
    const float* brow1 = &w1t[(1 * 16 + l15) * W1T_STRIDE + 2 * hi];
    const float* brow2 = &w1t[(2 * 16 + l15) * W1T_STRIDE + 2 * hi];
    const float* brow3 = &w1t[(3 * 16 + l15) * W1T_STRIDE + 2 * hi];

    #pragma unroll 4
    for (int k = 0; k < D_; k += 4) {
        v2f a  = *(const v2f*)(arow + k);
        v2f b0 = *(const v2f*)(brow0 + k);
        v2f b1f = *(const v2f*)(brow1 + k);
        v2f b2f = *(const v2f*)(brow2 + k);
        v2f b3 = *(const v2f*)(brow3 + k);
        acc[0] = __builtin_amdgcn_wmma_f32_16x16x4_f32(false, a, false, b0,  (short)0, acc[0], false, false);
        acc[1] = __builtin_amdgcn_wmma_f32_16x16x4_f32(false, a, false, b1f, (short)0, acc[1], false, false);
        acc[2] = __builtin_amdgcn_wmma_f32_16x16x4_f32(false, a, false, b2f, (short)0, acc[2], false, false);
        acc[3] = __builtin_amdgcn_wmma_f32_16x16x4_f32(false, a, false, b3,  (short)0, acc[3], false, false);
    }

    // Bias + ReLU in C-layout (col n = nt*16 + l15 fixed per lane), then
    // accumulate the W2 dot per row.
    float s[8];
    #pragma unroll
    for (int r = 0; r < 8; ++r) s[r] = 0.0f;
    #pragma unroll
    for (int nt = 0; nt < 4; ++nt) {
        const float bias = b1[nt * 16 + l15];
        const float w2v  = W2[nt * 16 + l15];
        #pragma unroll
        for (int r = 0; r < 8; ++r) {
            float h = acc[nt][r] + bias;
            h = h > 0.0f ? h : 0.0f;
            s[r] += h * w2v;
        }
    }

    // Reduce across the 16-lane half (rows 0..7 in lanes 0-15, rows 8..15 in 16-31)
    #pragma unroll
    for (int r = 0; r < 8; ++r) {
        float v = s[r];
        v += __shfl_xor(v, 8, 32);
        v += __shfl_xor(v, 4, 32);
        v += __shfl_xor(v, 2, 32);
        v += __shfl_xor(v, 1, 32);
        s[r] = v;
    }

    if (l15 == 0) {
        const float bb = b2[0];
        #pragma unroll
        for (int r = 0; r < 8; ++r) {
            const int t  = t0 + hi * 8 + r;
            const float iv = 1.0f / (1.0f + __expf(-(s[r] + bb)));  // in (0,1)
            imp[t] = iv;
            const int seg = (t >> 15) * C_ + ca[t];   // t>>15 == batch (N=32768)
            // iv > 0, so uint compare == float compare
            atomicMax(&segmaxU[seg], __float_as_uint(iv));
        }
    }
}

// ---------------------------------------------------------------------------
// Kernel 2: per-segment softmax denominator
// ---------------------------------------------------------------------------
__global__ __launch_bounds__(256) void denom_kernel(
    const float* __restrict__ imp, const int* __restrict__ ca,
    const float* __restrict__ segmax, float* __restrict__ denom)
{
    const int t = blockIdx.x * 256 + threadIdx.x;
    const int seg = (t >> 15) * C_ + ca[t];
    const float e = __expf(imp[t] - segmax[seg]);
    atomicAdd(&denom[seg], e);
}

// ---------------------------------------------------------------------------
// Kernel 3: softmax-weighted segment sum of features, accumulated in LDS
// (64 clusters x 128 dims per block), then one global atomic flush per block.
// ---------------------------------------------------------------------------
#define TOKS3 512
__global__ __launch_bounds__(256) void agg_kernel(
    const float* __restrict__ feat, const int* __restrict__ ca,
    const float* __restrict__ imp, const float* __restrict__ segmax,
    const float* __restrict__ denom, float* __restrict__ out)
{
    __shared__ float accs[C_ * D_];  // 32 KB
    __shared__ float ws_[TOKS3];
    __shared__ int   cs_[TOKS3];

    const int b     = blockIdx.x >> 6;       // 64 blocks per batch
    const int chunk = blockIdx.x & 63;
    const int base  = b * N_ + chunk * TOKS3;

    for (int i = threadIdx.x; i < C_ * D_; i += 256) accs[i] = 0.0f;
    for (int i = threadIdx.x; i < TOKS3; i += 256) {
        const int t   = base + i;
        const int c   = ca[t];
        const int seg = b * C_ + c;
        ws_[i] = __expf(imp[t] - segmax[seg]) / denom[seg];
        cs_[i] = c;
    }
    __syncthreads();

    const int d   = threadIdx.x & (D_ - 1);
    const int sub = threadIdx.x >> 7;        // 0..1
    for (int i = sub; i < TOKS3; i += 2) {
        const float v = feat[(size_t)(base + i) * D_ + d] * ws_[i];
        atomicAdd(&accs[cs_[i] * D_ + d], v);   // ds_add_f32, conflict-free per wave
    }
    __syncthreads();

    for (int i = threadIdx.x; i < C_ * D_; i += 256)
        atomicAdd(&out[(size_t)b * C_ * D_ + i], accs[i]);
}

// ---------------------------------------------------------------------------
extern "C" void kernel_launch(void* const* d_in, const int* in_sizes, int n_in,
                              void* d_out, int out_size, void* d_ws, size_t ws_size,
                              hipStream_t stream) {
    const float* feat = (const float*)d_in[0];
    const int*   ca   = (const int*)d_in[1];
    const float* W1   = (const float*)d_in[2];
    const float* b1   = (const float*)d_in[3];
    const float* W2   = (const float*)d_in[4];
    const float* b2   = (const float*)d_in[5];
    float* out = (float*)d_out;

    float* wsf    = (float*)d_ws;
    float* imp    = wsf;                 // BN floats
    float* segmax = wsf + BN_;           // B*C floats (uint-maxed bit patterns)
    float* denom  = wsf + BN_ + B_ * C_; // B*C floats

    init_kernel<<<(B_ * C_ * D_ + 255) / 256, 256, 0, stream>>>(out, segmax, denom);
    imp_kernel<<<BN_ / 64, 128, 0, stream>>>(feat, ca, W1, b1, W2, b2,
                                             imp, (unsigned int*)segmax);
    denom_kernel<<<BN_ / 256, 256, 0, stream>>>(imp, ca, segmax, denom);
    agg_kernel<<<B_ * (N_ / TOKS3), 256, 0, stream>>>(feat, ca, imp, segmax, denom, out);
}